// InsideLoss2D_9758165696608
// MI455X (gfx1250) — compile-verified
//
#include <hip/hip_runtime.h>
#include <hip/hip_bf16.h>

// ---------------- problem constants (from reference) ----------------
#define BB      8
#define MM      100
#define NN      50000
#define DD      3
#define N_ITP   10
#define QQ      (MM * N_ITP)          // 1000 queries per batch
#define QT      ((QQ + 15) / 16)      // 63 query tiles of 16
#define QP      ((QT + 1) / 2)        // 32 query-tile PAIRS per batch
#define NT      (NN / 16)             // 3125 shape tiles of 16 (exact)
#define NSPLIT  8                     // N-dimension splits for parallelism
#define TPS     ((NT + NSPLIT - 1) / NSPLIT)  // 391 tiles per split
#define EPSILON 0.01f

typedef __attribute__((ext_vector_type(2))) float v2f;
typedef __attribute__((ext_vector_type(8))) float v8f;

// Build the A fragment (16x4 f32) for one 16-query tile:
//   row m = (-2*qx, -2*qy, -2*qz, |q|^2)
// A layout: lanes 0-15 hold K=0,1 in v0,v1; lanes 16-31 hold K=2,3 (M=lane&15).
static __device__ __forceinline__
v2f make_afrag(const float* __restrict__ cage, int b, int qt, int lane)
{
    const int m = lane & 15;
    int g = qt * 16 + m;
    if (g >= QQ) g = QQ - 1;                      // pad tile with last query
    const int   ci  = g / N_ITP;
    const int   cj  = g - ci * N_ITP;
    const float t   = (float)cj * (1.0f / (float)(N_ITP - 1));
    const float omt = 1.0f - t;
    const int   ci1 = (ci + 1 == MM) ? 0 : ci + 1;   // roll(-1)
    const float* c0 = cage + ((size_t)b * MM + ci ) * DD;
    const float* c1 = cage + ((size_t)b * MM + ci1) * DD;
    const float qx = t * c1[0] + omt * c0[0];
    const float qy = t * c1[1] + omt * c0[1];
    const float qz = t * c1[2] + omt * c0[2];
    const float q2 = qx * qx + qy * qy + qz * qz;
    v2f a;
    if (lane < 16) { a.x = -2.0f * qx; a.y = -2.0f * qy; }   // K=0,1
    else           { a.x = -2.0f * qz; a.y = q2;         }   // K=2,3
    return a;
}

// ---------------------------------------------------------------------------
// Kernel 1: per-(batch, query-tile-pair, N-split) NN partial argmin.
// One wave per block.  V_WMMA_F32_16X16X4_F32 computes a 16x16 tile of exact
// squared distances:  D = A x B + C  with
//   A row m = (-2q, |q|^2),  B col n = (s, 1),  C[m][n] = |s_n|^2.
// Each loaded shape tile feeds TWO WMMAs (two query tiles) and the N loop is
// unrolled x2 so two b96 loads are in flight per iteration.
// ---------------------------------------------------------------------------
__global__ __launch_bounds__(32)
void nn_partial_kernel(const float* __restrict__ cage,
                       const float* __restrict__ shape,
                       float* __restrict__ d2_part,
                       int*   __restrict__ idx_part)
{
    const int lane = threadIdx.x;                 // 0..31 (wave32)
    int bs = blockIdx.x;
    const int ns = bs % NSPLIT;  bs /= NSPLIT;
    const int qp = bs % QP;      bs /= QP;
    const int b  = bs;                            // 0..7

    const int qt0 = qp * 2;
    const int qt1 = (qt0 + 1 < QT) ? qt0 + 1 : QT - 1;   // clamp (dup ok)

    const v2f a0 = make_afrag(cage, b, qt0, lane);
    const v2f a1 = make_afrag(cage, b, qt1, lane);

    const float* sb = shape + (size_t)b * NN * DD;
    const int t0 = ns * TPS;
    const int t1 = (t0 + TPS < NT) ? (t0 + TPS) : NT;
    const int ncol = lane & 15;

    float best0[8], best1[8];
    int   bidx0[8], bidx1[8];
#pragma unroll
    for (int r = 0; r < 8; ++r) {
        best0[r] = 3.0e38f; bidx0[r] = 0;
        best1[r] = 3.0e38f; bidx1[r] = 0;
    }

    int nt = t0;
    // ---- main loop: 2 shape tiles, 4 WMMAs per iteration ----
    for (; nt + 2 <= t1; nt += 2) {
        const int nA = nt * 16 + ncol;
        const int nB = nA + 16;
        const float* spA = sb + (size_t)nA * DD;
        const float* spB = sb + (size_t)nB * DD;
        const float sxA = spA[0], syA = spA[1], szA = spA[2];
        const float sxB = spB[0], syB = spB[1], szB = spB[2];
        __builtin_prefetch(spA + 32 * DD, 0, 1);   // 2 tiles ahead

        const float s2A = sxA * sxA + syA * syA + szA * szA;
        const float s2B = sxB * sxB + syB * syB + szB * szB;

        v2f bfA, bfB;
        if (lane < 16) { bfA.x = sxA; bfA.y = syA; bfB.x = sxB; bfB.y = syB; }
        else           { bfA.x = szA; bfA.y = 1.0f; bfB.x = szB; bfB.y = 1.0f; }

        v8f cA, cB;
#pragma unroll
        for (int r = 0; r < 8; ++r) { cA[r] = s2A; cB[r] = s2B; }

        const v8f d0A = __builtin_amdgcn_wmma_f32_16x16x4_f32(
            false, a0, false, bfA, (short)0, cA, false, false);
        const v8f d1A = __builtin_amdgcn_wmma_f32_16x16x4_f32(
            false, a1, false, bfA, (short)0, cA, false, false);
        const v8f d0B = __builtin_amdgcn_wmma_f32_16x16x4_f32(
            false, a0, false, bfB, (short)0, cB, false, false);
        const v8f d1B = __builtin_amdgcn_wmma_f32_16x16x4_f32(
            false, a1, false, bfB, (short)0, cB, false, false);

#pragma unroll
        for (int r = 0; r < 8; ++r) {
            bool lt;
            lt = d0A[r] < best0[r]; best0[r] = lt ? d0A[r] : best0[r]; bidx0[r] = lt ? nA : bidx0[r];
            lt = d1A[r] < best1[r]; best1[r] = lt ? d1A[r] : best1[r]; bidx1[r] = lt ? nA : bidx1[r];
            lt = d0B[r] < best0[r]; best0[r] = lt ? d0B[r] : best0[r]; bidx0[r] = lt ? nB : bidx0[r];
            lt = d1B[r] < best1[r]; best1[r] = lt ? d1B[r] : best1[r]; bidx1[r] = lt ? nB : bidx1[r];
        }
    }
    // ---- tail: one remaining shape tile (TPS is odd) ----
    if (nt < t1) {
        const int n = nt * 16 + ncol;
        const float* sp = sb + (size_t)n * DD;
        const float sx = sp[0], sy = sp[1], sz = sp[2];
        const float s2 = sx * sx + sy * sy + sz * sz;
        v2f bf;
        if (lane < 16) { bf.x = sx; bf.y = sy; }
        else           { bf.x = sz; bf.y = 1.0f; }
        v8f c;
#pragma unroll
        for (int r = 0; r < 8; ++r) c[r] = s2;
        const v8f d0 = __builtin_amdgcn_wmma_f32_16x16x4_f32(
            false, a0, false, bf, (short)0, c, false, false);
        const v8f d1 = __builtin_amdgcn_wmma_f32_16x16x4_f32(
            false, a1, false, bf, (short)0, c, false, false);
#pragma unroll
        for (int r = 0; r < 8; ++r) {
            bool lt;
            lt = d0[r] < best0[r]; best0[r] = lt ? d0[r] : best0[r]; bidx0[r] = lt ? n : bidx0[r];
            lt = d1[r] < best1[r]; best1[r] = lt ? d1[r] : best1[r]; bidx1[r] = lt ? n : bidx1[r];
        }
    }

    // ---- branchless argmin across N = across each 16-lane half ----
#pragma unroll
    for (int off = 1; off < 16; off <<= 1) {
#pragma unroll
        for (int r = 0; r < 8; ++r) {
            float od; int oi; bool take;
            od = __shfl_xor(best0[r], off, 32);
            oi = __shfl_xor(bidx0[r], off, 32);
            take = (od < best0[r]) | ((od == best0[r]) & (oi < bidx0[r]));
            best0[r] = take ? od : best0[r];
            bidx0[r] = take ? oi : bidx0[r];
            od = __shfl_xor(best1[r], off, 32);
            oi = __shfl_xor(bidx1[r], off, 32);
            take = (od < best1[r]) | ((od == best1[r]) & (oi < bidx1[r]));
            best1[r] = take ? od : best1[r];
            bidx1[r] = take ? oi : bidx1[r];
        }
    }

    // lane 0 holds M rows 0..7, lane 16 holds M rows 8..15
    if ((lane & 15) == 0) {
        const int mbase = (lane >> 4) * 8;
#pragma unroll
        for (int r = 0; r < 8; ++r) {
            const size_t o0 =
                ((((size_t)b * QT + qt0) * 16) + (mbase + r)) * NSPLIT + ns;
            d2_part[o0]  = best0[r];
            idx_part[o0] = bidx0[r];
            const size_t o1 =
                ((((size_t)b * QT + qt1) * 16) + (mbase + r)) * NSPLIT + ns;
            d2_part[o1]  = best1[r];
            idx_part[o1] = bidx1[r];
        }
    }
}

// ---------------------------------------------------------------------------
// Kernel 2: merge the NSPLIT partial argmins per query, gather nn point and
// normal, compute per-query loss = relu(-( (q - p - eps*n) . n )).
// ---------------------------------------------------------------------------
__global__ __launch_bounds__(128)
void merge_loss_kernel(const float* __restrict__ cage,
                       const float* __restrict__ shape,
                       const float* __restrict__ normals,
                       const float* __restrict__ d2_part,
                       const int*   __restrict__ idx_part,
                       float* __restrict__ loss_arr)
{
    const int qs = blockIdx.x * blockDim.x + threadIdx.x;
    if (qs >= BB * QT * 16) return;

    const int m  = qs & 15;
    const int qt = (qs >> 4) % QT;
    const int b  = qs / (16 * QT);
    const int g  = qt * 16 + m;
    if (g >= QQ) { loss_arr[qs] = 0.0f; return; }   // padded query slot

    // merge the NSPLIT partials (branchless, tie -> lower index)
    const size_t base = (size_t)qs * NSPLIT;
    float bd = d2_part[base];
    int   bi = idx_part[base];
#pragma unroll
    for (int s = 1; s < NSPLIT; ++s) {
        const float d  = d2_part[base + s];
        const int   ix = idx_part[base + s];
        const bool take = (d < bd) | ((d == bd) & (ix < bi));
        bd = take ? d  : bd;
        bi = take ? ix : bi;
    }

    // recompute interpolated cage point
    const int   ci  = g / N_ITP;
    const int   cj  = g - ci * N_ITP;
    const float t   = (float)cj * (1.0f / (float)(N_ITP - 1));
    const float omt = 1.0f - t;
    const int   ci1 = (ci + 1 == MM) ? 0 : ci + 1;
    const float* c0 = cage + ((size_t)b * MM + ci ) * DD;
    const float* c1 = cage + ((size_t)b * MM + ci1) * DD;
    const float qx = t * c1[0] + omt * c0[0];
    const float qy = t * c1[1] + omt * c0[1];
    const float qz = t * c1[2] + omt * c0[2];

    const float* p  = shape   + ((size_t)b * NN + bi) * DD;
    const float* nr = normals + ((size_t)b * NN + bi) * DD;
    const float nx = nr[0], ny = nr[1], nz = nr[2];
    const float vx = qx - p[0] - EPSILON * nx;
    const float vy = qy - p[1] - EPSILON * ny;
    const float vz = qz - p[2] - EPSILON * nz;
    const float dot = vx * nx + vy * ny + vz * nz;
    loss_arr[qs] = (dot < 0.0f) ? -dot : 0.0f;
}

// ---------------------------------------------------------------------------
// Kernel 3: single-block reduction of the per-query losses -> mean.
// ---------------------------------------------------------------------------
__global__ __launch_bounds__(256)
void reduce_kernel(const float* __restrict__ loss_arr, float* __restrict__ out)
{
    __shared__ float sm[256];
    const int tid = threadIdx.x;
    float s = 0.0f;
    for (int i = tid; i < BB * QT * 16; i += 256) s += loss_arr[i];
    sm[tid] = s;
    __syncthreads();
    for (int stride = 128; stride > 0; stride >>= 1) {
        if (tid < stride) sm[tid] += sm[tid + stride];
        __syncthreads();
    }
    if (tid == 0) out[0] = sm[0] * (1.0f / (float)(BB * QQ));
}

// ---------------------------------------------------------------------------
extern "C" void kernel_launch(void* const* d_in, const int* in_sizes, int n_in,
                              void* d_out, int out_size, void* d_ws, size_t ws_size,
                              hipStream_t stream)
{
    const float* cage    = (const float*)d_in[0];   // (8,100,3)
    const float* shape   = (const float*)d_in[1];   // (8,50000,3)
    const float* normals = (const float*)d_in[2];   // (8,50000,3)
    float* out = (float*)d_out;

    // scratch carve-up
    const size_t n_part = (size_t)BB * QT * 16 * NSPLIT;   // 64512
    char* ws = (char*)d_ws;
    float* d2_part  = (float*)ws;
    int*   idx_part = (int*)(ws + n_part * sizeof(float));
    float* loss_arr = (float*)(ws + 2 * n_part * sizeof(float));
    (void)ws_size; (void)in_sizes; (void)n_in; (void)out_size;

    // 1) partial nearest-neighbor argmin over N splits (WMMA distance tiles)
    nn_partial_kernel<<<BB * QP * NSPLIT, 32, 0, stream>>>(
        cage, shape, d2_part, idx_part);

    // 2) merge partials + per-query loss
    const int nq = BB * QT * 16;
    merge_loss_kernel<<<(nq + 127) / 128, 128, 0, stream>>>(
        cage, shape, normals, d2_part, idx_part, loss_arr);

    // 3) mean
    reduce_kernel<<<1, 256, 0, stream>>>(loss_arr, out);
}